// SemiMarkovHMM_5995774345473
// MI455X (gfx1250) — compile-verified
//
#include <hip/hip_runtime.h>
#include <hip/hip_bf16.h>
#include <stdint.h>

#define LOG2PI_F 1.8378770664093453f
#define NS   512    // NUM_STATES
#define DM   128    // OBS_DIM
#define BB   64     // batch
#define TT   2000   // frames
#define KK   400    // segments
#define TPB  125    // 16-frame tiles per batch (2000/16)
#define OBS_LDS_STRIDE 132  // 128 + 4 pad floats -> conflict-free fragment reads
#define SV_LDS_STRIDE  260  // 256 + 4 pad floats

typedef float  v2f_t  __attribute__((ext_vector_type(2)));
typedef float  v8f_t  __attribute__((ext_vector_type(8)));
typedef unsigned int v4u_t __attribute__((ext_vector_type(4)));
typedef int    v8i_t  __attribute__((ext_vector_type(8)));
typedef int    v4i_t  __attribute__((ext_vector_type(4)));

#if defined(__has_builtin)
#if __has_builtin(__builtin_amdgcn_tensor_load_to_lds) && __has_builtin(__builtin_amdgcn_s_wait_tensorcnt)
#define HAVE_TDM 1
#endif
#if __has_builtin(__builtin_amdgcn_wmma_f32_16x16x4_f32)
#define HAVE_WMMA_F32 1
#endif
#endif

#if __has_include(<hip/amd_detail/amd_gfx1250_TDM.h>)
#define TDM_6ARG 1
#else
#define TDM_6ARG 0
#endif

__device__ __forceinline__ float softplusf(float x) {
    return fmaxf(x, 0.0f) + log1pf(expf(-fabsf(x)));
}

// ---------------------------------------------------------------------------
// Kernel A: per-state tables.
//   svec[s][0:128]   = -2 * mu * w   (w = exp(-lv))
//   svec[s][128:256] =  w
//   sconst[s]        = sum_d (mu^2*w + lv + log2pi)
//   duration coefs: A1 = alpha-1, B = beta, C = lgamma(alpha) - alpha*log(beta)
// ---------------------------------------------------------------------------
__global__ __launch_bounds__(128) void k_state_tables(
    const float* __restrict__ alphaP, const float* __restrict__ betaP,
    const float* __restrict__ mu, const float* __restrict__ lv,
    float* __restrict__ svec, float* __restrict__ sconst,
    float* __restrict__ dA1, float* __restrict__ dB, float* __restrict__ dC)
{
    __shared__ float red[128];
    const int s = blockIdx.x;
    const int d = threadIdx.x;
    const float l = lv[s * DM + d];
    const float w = expf(-l);
    const float m = mu[s * DM + d];
    svec[(size_t)s * 256 + d]        = -2.0f * m * w;
    svec[(size_t)s * 256 + 128 + d]  = w;
    red[d] = m * m * w + l + LOG2PI_F;
    __syncthreads();
    for (int k = 64; k > 0; k >>= 1) {
        if (d < k) red[d] += red[d + k];
        __syncthreads();
    }
    if (d == 0) sconst[s] = red[0];
    if (d == 1) {
        const float alpha = softplusf(alphaP[s]) + 1e-6f;
        const float beta  = softplusf(betaP[s])  + 1e-6f;
        dA1[s] = alpha - 1.0f;
        dB[s]  = beta;
        dC[s]  = lgammaf(alpha) - alpha * logf(beta);
    }
}

// ---------------------------------------------------------------------------
// Kernel B: logsumexp of each transition row (blocks 0..511) and of the
// initial logits (block 512) -> lse[513].
// ---------------------------------------------------------------------------
__global__ __launch_bounds__(256) void k_lse(
    const float* __restrict__ trans, const float* __restrict__ initL,
    float* __restrict__ lse)
{
    __shared__ float red[256];
    const float* src = (blockIdx.x < NS) ? (trans + (size_t)blockIdx.x * NS) : initL;
    const int t = threadIdx.x;
    float m = -INFINITY;
    for (int i = t; i < NS; i += 256) m = fmaxf(m, src[i]);
    red[t] = m; __syncthreads();
    for (int k = 128; k > 0; k >>= 1) {
        if (t < k) red[t] = fmaxf(red[t], red[t + k]);
        __syncthreads();
    }
    const float mx = red[0];
    __syncthreads();
    float sum = 0.0f;
    for (int i = t; i < NS; i += 256) sum += expf(src[i] - mx);
    red[t] = sum; __syncthreads();
    for (int k = 128; k > 0; k >>= 1) {
        if (t < k) red[t] += red[t + k];
        __syncthreads();
    }
    if (t == 0) lse[blockIdx.x] = mx + logf(red[0]);
}

// ---------------------------------------------------------------------------
// Kernel C: per-batch bookkeeping. One block per batch:
//   - inclusive cumsum of durations, searchsorted(right) -> frame states
//   - duration log-prob sum, transition log-prob sum, initial log-prob
// ---------------------------------------------------------------------------
__global__ __launch_bounds__(256) void k_batch(
    const int* __restrict__ seq, const int* __restrict__ dur,
    const float* __restrict__ trans, const float* __restrict__ lse,
    const float* __restrict__ initL,
    const float* __restrict__ dA1, const float* __restrict__ dB,
    const float* __restrict__ dC,
    int* __restrict__ fstate, float* __restrict__ base)
{
    __shared__ int   scum[KK];
    __shared__ float red[256];
    const int b = blockIdx.x;
    const int t = threadIdx.x;
    const int* dseq = dur + b * KK;
    const int* sseq = seq + b * KK;
    if (t == 0) {
        int acc = 0;
        for (int k = 0; k < KK; ++k) { acc += dseq[k]; scum[k] = acc; }
    }
    __syncthreads();
    // frame states via searchsorted(cum, t, 'right'), clipped to K-1
    for (int f = t; f < TT; f += 256) {
        int lo = 0, hi = KK;
        while (lo < hi) {
            const int mid = (lo + hi) >> 1;
            if (scum[mid] <= f) lo = mid + 1; else hi = mid;
        }
        const int idx = (lo > KK - 1) ? (KK - 1) : lo;
        fstate[b * TT + f] = sseq[idx];
    }
    float lsum = 0.0f;
    for (int k = t; k < KK; k += 256) {           // gamma duration log-prob
        const int s = sseq[k];
        const float d = (float)dseq[k];
        const float lp = dA1[s] * logf(d + 1e-8f) - dB[s] * d - dC[s];
        lsum += (d >= 1.0f) ? lp : -INFINITY;
    }
    for (int k = t; k < KK - 1; k += 256) {       // transition log-prob
        const int s0 = sseq[k], s1 = sseq[k + 1];
        lsum += trans[(size_t)s0 * NS + s1] - lse[s0];
    }
    red[t] = lsum; __syncthreads();
    for (int k = 128; k > 0; k >>= 1) {
        if (t < k) red[t] += red[t + k];
        __syncthreads();
    }
    if (t == 0) base[b] = red[0] + (initL[sseq[0]] - lse[NS]);
}

// ---------------------------------------------------------------------------
// TDM: stage one contiguous 16x128 f32 obs tile (8 KB) into LDS with a
// 4-dword pad every 128 dwords (rows land at stride 132 floats).
// ---------------------------------------------------------------------------
#if defined(HAVE_TDM)
__device__ __forceinline__ void tdm_load_obs_tile(const float* gsrc, float* ldst) {
    const uint64_t ga = (uint64_t)(uintptr_t)gsrc;
    const uint32_t la = (uint32_t)(uintptr_t)ldst;  // workgroup-relative LDS offset
    v4u_t g0;
    g0.x = 1u;                                     // count=1 (valid user descriptor)
    g0.y = la;                                     // lds_addr
    g0.z = (uint32_t)ga;                           // global_addr[31:0]
    g0.w = (uint32_t)((ga >> 32) & 0x01FFFFFFu) | (2u << 30);  // addr[56:32], type=2
    v8i_t g1;
    // data_size=4B(2), pad_enable, pad_interval=128 dwords(6), pad_amount=4 dwords(3)
    g1[0] = (2 << 16) | (1 << 20) | (6 << 22) | (3 << 25);
    g1[1] = (int)(2048u << 16);                    // tensor_dim0 = 2048 (low 16 -> [63:48])
    g1[2] = (int)(1u << 16);                       // tensor_dim0 hi = 0; tensor_dim1 = 1
    g1[3] = (int)(2048u << 16);                    // tensor_dim1 hi = 0; tile_dim0 = 2048
    g1[4] = 1;                                     // tile_dim1 = 1, tile_dim2 = 0
    g1[5] = 2048;                                  // tensor_dim0_stride = 2048
    g1[6] = 0;
    g1[7] = 0;
    const v4i_t g2 = {0, 0, 0, 0};
    const v4i_t g3 = {0, 0, 0, 0};
#if TDM_6ARG
    const v8i_t g4 = {0, 0, 0, 0, 0, 0, 0, 0};
    __builtin_amdgcn_tensor_load_to_lds(g0, g1, g2, g3, g4, 0);
#else
    __builtin_amdgcn_tensor_load_to_lds(g0, g1, g2, g3, 0);
#endif
}
#endif

// ---------------------------------------------------------------------------
// Kernel D: the hot loop. One wave per 16-frame tile, 2 waves per block.
//   A (16x256) = [x, x^2] per frame, B (256x16) = gathered [-2*mu*w, w] per
//   frame's state. 64 chained V_WMMA_F32_16X16X4_F32; diagonal of D is the
//   quadratic form. tilep[tile] = sum over 16 frames of -0.5*(qf + c_s).
// ---------------------------------------------------------------------------
__global__ __launch_bounds__(64) void k_obs_wmma(
    const float* __restrict__ obs, const float* __restrict__ svec,
    const float* __restrict__ sconst, const int* __restrict__ fstate,
    float* __restrict__ tilep)
{
    __shared__ __attribute__((aligned(16))) float s_obs[2 * 16 * OBS_LDS_STRIDE];
    __shared__ __attribute__((aligned(16))) float s_sv [2 * 16 * SV_LDS_STRIDE];
    const int wv   = threadIdx.x >> 5;
    const int lane = threadIdx.x & 31;
    const int tile = blockIdx.x * 2 + wv;
    const int b    = tile / TPB;
    const int t0   = (tile % TPB) * 16;
    float* lobs = s_obs + wv * 16 * OBS_LDS_STRIDE;
    float* lsv  = s_sv  + wv * 16 * SV_LDS_STRIDE;
    const float* gobs = obs + ((size_t)b * TT + t0) * DM;

    // ---- stage obs tile (contiguous 8 KB) into LDS, rows at stride 132 ----
#if defined(HAVE_TDM)
    tdm_load_obs_tile(gobs, lobs);
#else
    {
        const float4* gsrc = reinterpret_cast<const float4*>(gobs);
        for (int i = lane; i < 512; i += 32) {     // 512 float4 per tile
            const float4 v = gsrc[i];
            const int row = i >> 5;                // 32 float4 per row
            const int c4  = i & 31;
            *reinterpret_cast<float4*>(lobs + row * OBS_LDS_STRIDE + c4 * 4) = v;
        }
    }
#endif

    // ---- gather per-frame state vectors into LDS columns ----
    int myState = 0;
    if (lane < 16) myState = fstate[b * TT + t0 + lane];
    for (int j = 0; j < 16; ++j) {
        const int sj = __shfl(myState, j, 32);
        const float4* src = reinterpret_cast<const float4*>(svec + (size_t)sj * 256);
        float4* dcol = reinterpret_cast<float4*>(lsv + j * SV_LDS_STRIDE);
        dcol[lane]      = src[lane];               // 64 float4 per column / 32 lanes
        dcol[lane + 32] = src[lane + 32];
    }
#if defined(HAVE_TDM)
    __builtin_amdgcn_s_wait_tensorcnt(0);
#endif
    __syncthreads();

    const int row  = lane & 15;                    // M for A, N for B
    const int koff = (lane >> 4) << 1;             // lanes 16-31 hold K+2,K+3
    const float* arow = lobs + row * OBS_LDS_STRIDE;
    const float* brow = lsv  + row * SV_LDS_STRIDE;
    float contrib;

#if defined(HAVE_WMMA_F32)
    v8f_t acc = {0.f, 0.f, 0.f, 0.f, 0.f, 0.f, 0.f, 0.f};
#pragma unroll
    for (int kk = 0; kk < 32; ++kk) {              // K 0..127: x vs -2*mu*w
        const int k0 = kk * 4 + koff;
        const float2 av = *reinterpret_cast<const float2*>(arow + k0);
        const float2 bv = *reinterpret_cast<const float2*>(brow + k0);
        v2f_t a;  a.x = av.x;  a.y = av.y;
        v2f_t bb; bb.x = bv.x; bb.y = bv.y;
        acc = __builtin_amdgcn_wmma_f32_16x16x4_f32(
            false, a, false, bb, (short)0, acc, false, false);
    }
#pragma unroll
    for (int kk = 32; kk < 64; ++kk) {             // K 128..255: x^2 vs w
        const int k0 = kk * 4 + koff;
        const float2 av = *reinterpret_cast<const float2*>(arow + (k0 - 128));
        const float2 bv = *reinterpret_cast<const float2*>(brow + k0);
        v2f_t a;  a.x = av.x * av.x;  a.y = av.y * av.y;
        v2f_t bb; bb.x = bv.x; bb.y = bv.y;
        acc = __builtin_amdgcn_wmma_f32_16x16x4_f32(
            false, a, false, bb, (short)0, acc, false, false);
    }
    // Diagonal of D: element (i,i) lives in VGPR i @ lane i (i<8) and
    // VGPR i-8 @ lane 16+i (i>=8).
    contrib = (lane < 16) ? (-0.5f * sconst[myState]) : 0.0f;
    float diag = 0.0f;
#pragma unroll
    for (int r = 0; r < 8; ++r) {
        if (lane == r)      diag = acc[r];
        if (lane == 24 + r) diag = acc[r];
    }
    contrib += -0.5f * diag;
#else
    // scalar fallback: lanes 0-15 compute their own frame's dot product
    float qf = 0.0f;
    if (lane < 16) {
        for (int k = 0; k < 128; ++k) {
            const float x = arow[k];
            qf += x * brow[k] + x * x * brow[128 + k];
        }
    }
    contrib = (lane < 16) ? (-0.5f * (qf + sconst[myState])) : 0.0f;
#endif
#pragma unroll
    for (int off = 16; off > 0; off >>= 1)
        contrib += __shfl_down(contrib, off, 32);
    if (lane == 0) tilep[tile] = contrib;
}

// ---------------------------------------------------------------------------
// Kernel E: per-batch final reduction of 125 tile partials + base terms.
// ---------------------------------------------------------------------------
__global__ __launch_bounds__(128) void k_final(
    const float* __restrict__ tilep, const float* __restrict__ base,
    float* __restrict__ out)
{
    __shared__ float red[128];
    const int b = blockIdx.x, t = threadIdx.x;
    red[t] = (t < TPB) ? tilep[b * TPB + t] : 0.0f;
    __syncthreads();
    for (int k = 64; k > 0; k >>= 1) {
        if (t < k) red[t] += red[t + k];
        __syncthreads();
    }
    if (t == 0) out[b] = red[0] + base[b];
}

// ---------------------------------------------------------------------------
extern "C" void kernel_launch(void* const* d_in, const int* in_sizes, int n_in,
                              void* d_out, int out_size, void* d_ws, size_t ws_size,
                              hipStream_t stream) {
    (void)in_sizes; (void)n_in; (void)out_size; (void)ws_size;
    const float* obs    = (const float*)d_in[0];
    const float* alphaP = (const float*)d_in[1];
    const float* betaP  = (const float*)d_in[2];
    const float* trans  = (const float*)d_in[3];
    const float* initL  = (const float*)d_in[4];
    const float* means  = (const float*)d_in[5];
    const float* lvars  = (const float*)d_in[6];
    const int*   seq    = (const int*)d_in[7];
    const int*   dur    = (const int*)d_in[8];
    float* out = (float*)d_out;

    char* ws = (char*)d_ws;
    float* svec   = (float*)(ws + 0);          // 512*256 f32  (512 KB)
    float* sconst = (float*)(ws + 524288);     // 512 f32
    float* dA1    = (float*)(ws + 526336);     // 512 f32
    float* dB     = (float*)(ws + 528384);     // 512 f32
    float* dC     = (float*)(ws + 530432);     // 512 f32
    float* lse    = (float*)(ws + 532480);     // 513 f32
    int*   fstate = (int*)  (ws + 534784);     // 64*2000 i32 (512 KB)
    float* base   = (float*)(ws + 1046784);    // 64 f32
    float* tilep  = (float*)(ws + 1047040);    // 64*125 f32

    k_state_tables<<<NS, 128, 0, stream>>>(alphaP, betaP, means, lvars,
                                           svec, sconst, dA1, dB, dC);
    k_lse<<<NS + 1, 256, 0, stream>>>(trans, initL, lse);
    k_batch<<<BB, 256, 0, stream>>>(seq, dur, trans, lse, initL,
                                    dA1, dB, dC, fstate, base);
    k_obs_wmma<<<(BB * TPB) / 2, 64, 0, stream>>>(obs, svec, sconst, fstate, tilep);
    k_final<<<BB, 128, 0, stream>>>(tilep, base, out);
}